// DualModeSinkhorn_4904852652233
// MI455X (gfx1250) — compile-verified
//
#include <hip/hip_runtime.h>
#include <math.h>

#define NS    8
#define BB    32
#define HH    128
#define WW    128
#define HWSZ  (HH * WW)          // 16384
#define PQ    (HWSZ / 4)         // 4096 float4 per channel-plane
#define CHN   (BB * NS)          // 256 marginal channels
#define EPSF  1e-8f

typedef __attribute__((ext_vector_type(2))) float v2f;
typedef __attribute__((ext_vector_type(4))) float f4;
typedef __attribute__((ext_vector_type(8))) float v8f;

#if __has_builtin(__builtin_amdgcn_global_load_async_to_lds_b32) && \
    __has_builtin(__builtin_amdgcn_s_wait_asynccnt)
#define USE_ASYNC_LDS 1
typedef __attribute__((address_space(1))) int gint_t;   // global int*
typedef __attribute__((address_space(3))) int lint_t;   // LDS int*
#endif

// ---------------------------------------------------------------------------
// w1d[t] = sum_j k2d[t][j]  (exact 1D factor of the separable Gaussian)
__global__ void kprepw(const float* __restrict__ K, float* __restrict__ w1d) {
    int t = threadIdx.x;
    if (t < 5) {
        float s = 0.0f;
        for (int q = 0; q < 5; ++q) s += K[t * 5 + q];
        w1d[t] = s;
    }
}

// ---------------------------------------------------------------------------
// H0 = exp(log_in); r = c = 1.  One thread per (b,i,4-pixel group), loops j.
__global__ void kinit(const f4* __restrict__ logH, f4* __restrict__ H0,
                      f4* __restrict__ r, f4* __restrict__ c) {
    int t  = blockIdx.x * blockDim.x + threadIdx.x;   // < CHN*PQ
    int p  = t & (PQ - 1);
    int ch = t >> 12;                                 // b*8 + i
    size_t base = (size_t)ch * 8 * PQ + p;
#pragma unroll
    for (int j = 0; j < 8; ++j) {
        f4 v = logH[base + (size_t)j * PQ];
        f4 e;
        e.x = expf(v.x); e.y = expf(v.y); e.z = expf(v.z); e.w = expf(v.w);
        H0[base + (size_t)j * PQ] = e;
    }
    f4 one = {1.0f, 1.0f, 1.0f, 1.0f};
    r[t] = one;
    c[t] = one;
}

// ---------------------------------------------------------------------------
// Marginal of the *current* Hl = H0 * r ⊗ c, without materializing Hl.
// mode 0 (row, per i): M[b,i,p] = r[b,i,p] * sum_j H0[b,i,j,p] * c[b,j,p]
// mode 1 (col, per j): M[b,j,p] = c[b,j,p] * sum_i H0[b,i,j,p] * r[b,i,p]
// float4 along pixels: b128 loads, 512B per wave per vmem instruction.
__global__ void kmarg(const f4* __restrict__ H0,
                      const f4* __restrict__ selfS,
                      const f4* __restrict__ otherS,
                      f4* __restrict__ Mout, int mode) {
    int t  = blockIdx.x * blockDim.x + threadIdx.x;   // < CHN*PQ
    int p  = t & (PQ - 1);
    int ch = t >> 12;                                 // b*8 + cc
    int b  = ch >> 3;
    int cc = ch & 7;
    const f4* o = otherS + (size_t)(b * 8) * PQ + p;
    f4 s = {0.0f, 0.0f, 0.0f, 0.0f};
    if (mode == 0) {
        const f4* h = H0 + (size_t)(b * 64 + cc * 8) * PQ + p;
#pragma unroll
        for (int j = 0; j < 8; ++j)
            s += h[(size_t)j * PQ] * o[(size_t)j * PQ];
    } else {
        const f4* h = H0 + (size_t)(b * 64 + cc) * PQ + p;
#pragma unroll
        for (int j = 0; j < 8; ++j)
            s += h[(size_t)j * 8 * PQ] * o[(size_t)j * PQ];
    }
    Mout[t] = selfS[t] * s;
}

// ---------------------------------------------------------------------------
// Separable 5x5 wrap conv of M via V_WMMA_F32_16X16X4_F32, fused with the
// Sinkhorn scale update:  scale[ch,y,x] /= (conv(M)[ch,y,x] + EPS).
// One wave = one 16x16 tile; 4 waves per block; LDS stages the 20x20 input
// window (async global->LDS when available) and the 20x16 horizontal result.
__global__ void __launch_bounds__(128)
ksmooth(const float* __restrict__ M, float* __restrict__ scale,
        const float* __restrict__ w1d) {
    __shared__ float sm[4][20 * 21 + 20 * 17];
    const int tid  = threadIdx.x;
    const int wave = tid >> 5;
    const int lane = tid & 31;
    float* in_s = &sm[wave][0];        // 20 rows x 21 stride (input window)
    float* h_s  = &sm[wave][420];      // 20 rows x 17 stride (horizontal result)

    const int tile = blockIdx.x * 4 + wave;        // 0..16383
    const int tx   = tile & 7;
    const int ty   = (tile >> 3) & 7;
    const int cidx = tile >> 6;                    // 0..255
    const int x0   = tx * 16;
    const int y0   = ty * 16;
    const float* Mc = M + (size_t)cidx * HWSZ;

    const int m16 = lane & 15;         // M index for A-weights, N index for B-weights
    const int k0  = (lane >> 4) * 2;   // K base per half-wave (ISA layout)

    // Shifted-weight operand W_g: value w1d[4g + k - m16] (0 outside [0,5)).
    // Identical per-lane content serves as B (horizontal) and A (vertical).
    v2f W5[5];
#pragma unroll
    for (int g = 0; g < 5; ++g) {
        int d0 = 4 * g + k0 - m16;
        int d1 = d0 + 1;
        W5[g].x = (d0 >= 0 && d0 < 5) ? w1d[d0] : 0.0f;
        W5[g].y = (d1 >= 0 && d1 < 5) ? w1d[d1] : 0.0f;
    }

    // Stage wrapped 20x20 window (image rows y0-2..y0+17, cols x0-2..x0+17).
    for (int e = lane; e < 400; e += 32) {
        int rr = e / 20, qq = e % 20;
        int yr = (y0 - 2 + rr) & (HH - 1);
        int xq = (x0 - 2 + qq) & (WW - 1);
#ifdef USE_ASYNC_LDS
        __builtin_amdgcn_global_load_async_to_lds_b32(
            (gint_t*)(uintptr_t)(const void*)(Mc + yr * WW + xq),
            (lint_t*)(unsigned)(uintptr_t)(void*)(&in_s[rr * 21 + qq]),
            0, 0);
#else
        in_s[rr * 21 + qq] = Mc[yr * WW + xq];
#endif
    }
#ifdef USE_ASYNC_LDS
    __builtin_amdgcn_s_wait_asynccnt(0);
#endif
    __syncthreads();

    // Horizontal pass: two overlapping 16-row blocks cover window rows 0..19.
    const int rowhalf = (lane >> 4) * 8;
#pragma unroll
    for (int rb = 0; rb < 2; ++rb) {
        const int rbase = rb * 4;
        const int r = rbase + m16;     // window row for this lane's A rows
        v8f acc = {};
#pragma unroll
        for (int g = 0; g < 5; ++g) {
            v2f A;
            A.x = in_s[r * 21 + 4 * g + k0];
            A.y = in_s[r * 21 + 4 * g + k0 + 1];
            acc = __builtin_amdgcn_wmma_f32_16x16x4_f32(
                false, A, false, W5[g], (short)0, acc, false, false);
        }
#pragma unroll
        for (int v = 0; v < 8; ++v) {
            int m  = v + rowhalf;      // row within block
            int wr = rbase + m;        // window row
            if (rb == 0 || wr >= 16)   // block 1 contributes rows 16..19 only
                h_s[wr * 17 + m16] = acc[v];
        }
    }
    __syncthreads();

    // Vertical pass: D = sum_g W_g(16x4) x Hw_g(4x16).
    v8f acc2 = {};
#pragma unroll
    for (int g = 0; g < 5; ++g) {
        v2f Bv;
        Bv.x = h_s[(4 * g + k0) * 17 + m16];
        Bv.y = h_s[(4 * g + k0 + 1) * 17 + m16];
        acc2 = __builtin_amdgcn_wmma_f32_16x16x4_f32(
            false, W5[g], false, Bv, (short)0, acc2, false, false);
    }

    // Fused scale update: scale /= (smoothed + eps)
    float* Sc = scale + (size_t)cidx * HWSZ;
#pragma unroll
    for (int v = 0; v < 8; ++v) {
        int y = y0 + v + rowhalf;
        int x = x0 + m16;
        size_t o = (size_t)y * WW + x;
        Sc[o] = Sc[o] / (acc2[v] + EPSF);
    }
}

// ---------------------------------------------------------------------------
// out[b,i,j,p] = exp(log(H0 * r[b,i,p] * c[b,j,p] + EPS))   (in place, float4)
__global__ void kfinal(f4* __restrict__ H0, const f4* __restrict__ r,
                       const f4* __restrict__ c) {
    size_t t = (size_t)blockIdx.x * blockDim.x + threadIdx.x;  // < out/4
    int p    = (int)(t & (PQ - 1));
    int ch64 = (int)(t >> 12);                 // b*64 + i*8 + j
    int j = ch64 & 7, i = (ch64 >> 3) & 7, b = ch64 >> 6;
    f4 v = H0[t] * r[(size_t)(b * 8 + i) * PQ + p]
                 * c[(size_t)(b * 8 + j) * PQ + p];
    f4 o;
    o.x = expf(logf(v.x + EPSF));
    o.y = expf(logf(v.y + EPSF));
    o.z = expf(logf(v.z + EPSF));
    o.w = expf(logf(v.w + EPSF));
    H0[t] = o;
}

// ---------------------------------------------------------------------------
extern "C" void kernel_launch(void* const* d_in, const int* in_sizes, int n_in,
                              void* d_out, int out_size, void* d_ws, size_t ws_size,
                              hipStream_t stream) {
    const f4*    logH = (const f4*)d_in[0];
    const float* Kg   = (const float*)d_in[1];
    f4*   H0 = (f4*)d_out;                     // d_out doubles as the H0 buffer
    char* ws = (char*)d_ws;

    const size_t ME = (size_t)CHN * HWSZ;      // 4,194,304 floats (16 MB)
    float* Mb  = (float*)ws;                   // marginal buffer
    float* r   = (float*)(ws + ME * 4);        // row scales  (per b,i,pixel)
    float* c   = (float*)(ws + ME * 8);        // col scales  (per b,j,pixel)
    float* w1d = (float*)(ws + ME * 12);       // 5 floats

    kprepw<<<1, 32, 0, stream>>>(Kg, w1d);

    const int tb    = 256;
    const int nMarg = (int)(ME / 4 / tb);      // 4096 blocks (float4 threads)
    kinit<<<nMarg, tb, 0, stream>>>(logH, H0, (f4*)r, (f4*)c);

    const int nSm = (CHN * 8 * 8) / 4;         // 16384 tiles, 4 waves/block
    for (int it = 0; it < 20; ++it) {
        kmarg<<<nMarg, tb, 0, stream>>>(H0, (f4*)r, (f4*)c, (f4*)Mb, 0);
        ksmooth<<<nSm, 128, 0, stream>>>(Mb, r, w1d);       // r /= (conv+eps)
        kmarg<<<nMarg, tb, 0, stream>>>(H0, (f4*)c, (f4*)r, (f4*)Mb, 1);
        ksmooth<<<nSm, 128, 0, stream>>>(Mb, c, w1d);       // c /= (conv+eps)
    }

    const int nFin = (int)((size_t)out_size / 4 / tb);      // 32768 blocks
    kfinal<<<nFin, tb, 0, stream>>>(H0, (f4*)r, (f4*)c);
}